// DecoderLSTM_49297634623946
// MI455X (gfx1250) — compile-verified
//
#include <hip/hip_runtime.h>

// ---------------------------------------------------------------------------
// DecoderLSTM (Show-Attend-Tell style) for MI455X gfx1250.
// GEMMs via V_WMMA_F32_16X16X4_F32 (fp32 WMMA == reference numerics).
// Weights are pre-transposed ONCE into a pair-interleaved layout:
//     P[(k/2)*2N + 2n + (k&1)] = W^T[k][n]
// so every B fragment is one contiguous global_load_b64 per lane (lanes 0-15
// and 16-31 each read a contiguous 128B segment). Weights (~70MB) stay
// resident in the 192MB L2 across all 51 time steps.
// The vocab GEMM wave carries all of M=64 (MT=4) to cut W_fc L2 traffic 4x;
// the small N=2048 GEMMs use MT=1 for maximum wave parallelism on the
// sequential critical path.
// ---------------------------------------------------------------------------

#define B_     64
#define P_     196
#define ENC_   2048
#define HID_   512
#define EMB_   300
#define VOCAB_ 20000
#define L_     52
#define T_     51
#define XH_    (EMB_ + ENC_ + HID_)   // 2860
#define G4_    (4 * HID_)             // 2048

typedef __attribute__((ext_vector_type(8))) float v8f;
typedef __attribute__((ext_vector_type(2))) float v2f;

__device__ __forceinline__ float sigf(float x) { return 1.0f / (1.0f + expf(-x)); }

// ------------------------- setup kernels -----------------------------------

// Stable descending argsort of caption lengths (B=64), O(n^2) rank trick.
__global__ void dl_sort_kernel(const int* __restrict__ lengths,
                               int* __restrict__ sort_ind,
                               int* __restrict__ dec_len,
                               float* __restrict__ out_declen,
                               float* __restrict__ out_sortind) {
  int i = threadIdx.x;
  if (i >= B_) return;
  int key = lengths[i];
  int r = 0;
  for (int j = 0; j < B_; ++j) {
    int kj = lengths[j];
    if (kj > key || (kj == key && j < i)) r++;
  }
  sort_ind[r]    = i;
  dec_len[r]     = key - 1;
  out_sortind[r] = (float)i;
  out_declen[r]  = (float)(key - 1);
}

__global__ void dl_caps_kernel(const int* __restrict__ caps,
                               const int* __restrict__ sort_ind,
                               float* __restrict__ out_caps) {
  int idx = blockIdx.x * blockDim.x + threadIdx.x;
  if (idx >= B_ * L_) return;
  int b = idx / L_, t = idx % L_;
  out_caps[idx] = (float)caps[sort_ind[b] * L_ + t];
}

__global__ void dl_encmean_kernel(const float* __restrict__ enc,
                                  const int* __restrict__ sort_ind,
                                  float* __restrict__ enc_mean) {
  int idx = blockIdx.x * blockDim.x + threadIdx.x;
  if (idx >= B_ * ENC_) return;
  int b = idx / ENC_, e = idx % ENC_;
  const float* p = enc + (size_t)sort_ind[b] * P_ * ENC_ + e;
  float s = 0.0f;
  for (int q = 0; q < P_; ++q) s += p[(size_t)q * ENC_];
  enc_mean[idx] = s * (1.0f / (float)P_);
}

__global__ void dl_emb_kernel(const float* __restrict__ embedding,
                              const int* __restrict__ caps,
                              const int* __restrict__ sort_ind,
                              float* __restrict__ emb_s) {
  int idx = blockIdx.x * blockDim.x + threadIdx.x;
  if (idx >= B_ * T_ * EMB_) return;
  int b = idx / (T_ * EMB_);
  int r = idx % (T_ * EMB_);
  int t = r / EMB_, e = r % EMB_;
  int tok = caps[sort_ind[b] * L_ + t];
  emb_s[idx] = embedding[(size_t)tok * EMB_ + e];
}

// Pair-interleaved transpose: out[(k/2)*2N + 2n + (k&1)] = in[n*K + k]
__global__ void dl_transpose_pack_kernel(const float* __restrict__ in,
                                         float* __restrict__ out, int N,
                                         int K) {
  size_t idx = (size_t)blockIdx.x * blockDim.x + threadIdx.x;
  if (idx >= (size_t)N * K) return;
  int k = (int)(idx / N), n = (int)(idx % N);
  out[(size_t)(k >> 1) * 2 * N + 2 * n + (k & 1)] = in[(size_t)n * K + k];
}

// Pair-interleaved [W_ih|W_hh]^T : k<2348 -> W_ih[n][k], else W_hh[n][k-2348]
__global__ void dl_wcat_pack_kernel(const float* __restrict__ W_ih,
                                    const float* __restrict__ W_hh,
                                    float* __restrict__ WcatP) {
  size_t idx = (size_t)blockIdx.x * blockDim.x + threadIdx.x;
  if (idx >= (size_t)XH_ * G4_) return;
  int k = (int)(idx / G4_), n = (int)(idx % G4_);
  float v = (k < EMB_ + ENC_)
                ? W_ih[(size_t)n * (EMB_ + ENC_) + k]
                : W_hh[(size_t)n * HID_ + (k - (EMB_ + ENC_))];
  WcatP[(size_t)(k >> 1) * 2 * G4_ + 2 * n + (k & 1)] = v;
}

__global__ void dl_init_kernel(const float* __restrict__ b_ih,
                               const float* __restrict__ b_hh,
                               float* __restrict__ bsum,
                               float* __restrict__ h, float* __restrict__ c) {
  int idx = blockIdx.x * blockDim.x + threadIdx.x;
  if (idx >= B_ * HID_) return;
  h[idx] = 0.0f;
  c[idx] = 0.0f;
  if (idx < G4_) bsum[idx] = b_ih[idx] + b_hh[idx];
}

// ------------------------- per-step kernels --------------------------------

// xh[:, 0:300] = e_t ; xh[:, 2348:2860] = h
__global__ void dl_buildx_kernel(const float* __restrict__ emb_s,
                                 const float* __restrict__ h,
                                 float* __restrict__ xh, int t) {
  int idx = blockIdx.x * blockDim.x + threadIdx.x;
  if (idx >= B_ * (EMB_ + HID_)) return;
  int b = idx / (EMB_ + HID_), j = idx % (EMB_ + HID_);
  if (j < EMB_)
    xh[(size_t)b * XH_ + j] = emb_s[((size_t)b * T_ + t) * EMB_ + j];
  else
    xh[(size_t)b * XH_ + (EMB_ + ENC_) + (j - EMB_)] = h[b * HID_ + (j - EMB_)];
}

__global__ void dl_cell_kernel(const float* __restrict__ gates,
                               const int* __restrict__ dec_len,
                               float* __restrict__ h, float* __restrict__ c,
                               float* __restrict__ h_new, int t) {
  int idx = blockIdx.x * blockDim.x + threadIdx.x;
  if (idx >= B_ * HID_) return;
  int b = idx / HID_, n = idx % HID_;
  const float* g = gates + (size_t)b * G4_;
  float i_ = sigf(g[n]);
  float f_ = sigf(g[n + HID_]);
  float gg = tanhf(g[n + 2 * HID_]);
  float o_ = sigf(g[n + 3 * HID_]);
  float cn = f_ * c[idx] + i_ * gg;
  float hn = o_ * tanhf(cn);
  h_new[idx] = hn;                 // preds use unmasked h_new (per reference)
  if (t < dec_len[b]) { h[idx] = hn; c[idx] = cn; }
}

// ------------------------- fp32 WMMA GEMM ----------------------------------
// C(M=64, N) = A(M, K) * B^T(K, N); each wave owns an (MT*16) x 32 tile.
// B is pair-interleaved (see above): one v2f load per fragment per lane.
// Fragment layouts per ISA 7.12.2:
//   A 16x4:  lanes 0-15 -> {K=k0,k0+1}, lanes 16-31 -> {K=k0+2,k0+3}, M=lane&15
//   B 4x16:  VGPR0 -> rows {k0 (lanes0-15), k0+2 (lanes16-31)}, VGPR1 -> +1
//   C/D:     VGPR j -> row j + 8*(lane>=16), col = lane&15
// MODE 0: v = sigmoid(acc+b_beta[n]) * enc_mean[m][n]  -> xh slice
// MODE 1: v = acc + bsum[n]                            -> gates
// MODE 2: v = (t < dec_len[m]) ? acc + b_fc[n] : 0     -> d_out slice
template <int MODE, int MT>
__global__ __launch_bounds__(256) void dl_wmma_gemm_kernel(
    const float* __restrict__ A, int lda, const float* __restrict__ Bpk, int N,
    int K, const float* __restrict__ bias, float* __restrict__ out, size_t ldo,
    const float* __restrict__ enc_mean, const int* __restrict__ dec_len,
    int t) {
  const int wave    = (int)((blockIdx.x * (size_t)blockDim.x + threadIdx.x) >> 5);
  const int nstrips = N >> 5;
  const int groups  = (B_ / 16) / MT;
  const int mtg     = wave / nstrips;
  if (mtg >= groups) return;
  const int ns   = wave - mtg * nstrips;
  const int lane = threadIdx.x & 31;
  const int half = lane >> 4;
  const int lm   = lane & 15;
  const int n0   = ns << 5;

  const float* Ap[MT];
#pragma unroll
  for (int mi = 0; mi < MT; ++mi)
    Ap[mi] = A + (size_t)((mtg * MT + mi) * 16 + lm) * lda + 2 * half;

  v8f acc[MT][2];
#pragma unroll
  for (int mi = 0; mi < MT; ++mi) {
    acc[mi][0] = {};
    acc[mi][1] = {};
  }

  const size_t ldp = 2 * (size_t)N;  // packed row-pair stride
  for (int k0 = 0; k0 < K; k0 += 4) {
    // B fragments: rows (k0+2*half, k0+2*half+1) -> packed pair (k0>>1)+half
    const float* Bp = Bpk + (size_t)((k0 >> 1) + half) * ldp + 2 * lm;
    v2f b0 = *(const v2f*)(Bp + 2 * n0);
    v2f b1 = *(const v2f*)(Bp + 2 * n0 + 32);
    v2f a[MT];
#pragma unroll
    for (int mi = 0; mi < MT; ++mi) a[mi] = *(const v2f*)(Ap[mi] + k0);
#pragma unroll
    for (int mi = 0; mi < MT; ++mi) {
      acc[mi][0] = __builtin_amdgcn_wmma_f32_16x16x4_f32(
          false, a[mi], false, b0, (short)0, acc[mi][0], false, false);
      acc[mi][1] = __builtin_amdgcn_wmma_f32_16x16x4_f32(
          false, a[mi], false, b1, (short)0, acc[mi][1], false, false);
    }
  }

#pragma unroll
  for (int mi = 0; mi < MT; ++mi) {
#pragma unroll
    for (int j = 0; j < 8; ++j) {
      int row = (mtg * MT + mi) * 16 + half * 8 + j;
#pragma unroll
      for (int s = 0; s < 2; ++s) {
        int col = n0 + s * 16 + lm;
        float v = acc[mi][s][j];
        if (MODE == 0) {
          v = sigf(v + bias[col]) * enc_mean[(size_t)row * N + col];
        } else if (MODE == 1) {
          v += bias[col];
        } else {
          v = (t < dec_len[row]) ? (v + bias[col]) : 0.0f;
        }
        out[(size_t)row * ldo + col] = v;
      }
    }
  }
}

// ------------------------- host launcher -----------------------------------

static inline int cdiv(long a, long b) { return (int)((a + b - 1) / b); }

extern "C" void kernel_launch(void* const* d_in, const int* in_sizes, int n_in,
                              void* d_out, int out_size, void* d_ws,
                              size_t ws_size, hipStream_t stream) {
  (void)in_sizes; (void)n_in; (void)out_size; (void)ws_size;

  const float* enc       = (const float*)d_in[0];
  const int*   caps      = (const int*)d_in[1];
  const int*   lens      = (const int*)d_in[2];
  const float* embedding = (const float*)d_in[3];
  const float* W_ih      = (const float*)d_in[4];
  const float* W_hh      = (const float*)d_in[5];
  const float* b_ih      = (const float*)d_in[6];
  const float* b_hh      = (const float*)d_in[7];
  const float* W_beta    = (const float*)d_in[8];
  const float* b_beta    = (const float*)d_in[9];
  const float* W_fc      = (const float*)d_in[10];
  const float* b_fc      = (const float*)d_in[11];

  float* out         = (float*)d_out;
  float* out_pred    = out;                                   // B*T*VOCAB
  float* out_caps    = out + (size_t)B_ * T_ * VOCAB_;        // B*L
  float* out_declen  = out_caps + (size_t)B_ * L_;            // B
  float* out_sortind = out_declen + B_;                       // B

  // ---- workspace carve (ints first, 512B, then 16B-aligned float regions)
  int*   sort_ind = (int*)d_ws;
  int*   dec_len  = sort_ind + 64;
  float* fw       = (float*)((char*)d_ws + 512);
  float* h        = fw;               fw += B_ * HID_;
  float* c        = fw;               fw += B_ * HID_;
  float* h_new    = fw;               fw += B_ * HID_;
  float* xh       = fw;               fw += (size_t)B_ * XH_;
  float* gates    = fw;               fw += (size_t)B_ * G4_;
  float* enc_mean = fw;               fw += (size_t)B_ * ENC_;
  float* emb_s    = fw;               fw += (size_t)B_ * T_ * EMB_;
  float* bsum     = fw;               fw += G4_;
  float* W_betaP  = fw;               fw += (size_t)HID_ * ENC_;   // packed
  float* WcatP    = fw;               fw += (size_t)XH_ * G4_;     // packed
  float* W_fcP    = fw;               fw += (size_t)HID_ * VOCAB_; // packed
  // total ~75 MB

  const int TPB = 256;

  // ---- setup
  dl_sort_kernel<<<1, 64, 0, stream>>>(lens, sort_ind, dec_len, out_declen,
                                       out_sortind);
  dl_caps_kernel<<<cdiv(B_ * L_, TPB), TPB, 0, stream>>>(caps, sort_ind,
                                                         out_caps);
  dl_encmean_kernel<<<cdiv((long)B_ * ENC_, TPB), TPB, 0, stream>>>(
      enc, sort_ind, enc_mean);
  dl_emb_kernel<<<cdiv((long)B_ * T_ * EMB_, TPB), TPB, 0, stream>>>(
      embedding, caps, sort_ind, emb_s);
  dl_transpose_pack_kernel<<<cdiv((long)ENC_ * HID_, TPB), TPB, 0, stream>>>(
      W_beta, W_betaP, ENC_, HID_);
  dl_transpose_pack_kernel<<<cdiv((long)VOCAB_ * HID_, TPB), TPB, 0, stream>>>(
      W_fc, W_fcP, VOCAB_, HID_);
  dl_wcat_pack_kernel<<<cdiv((long)XH_ * G4_, TPB), TPB, 0, stream>>>(
      W_ih, W_hh, WcatP);
  dl_init_kernel<<<cdiv(B_ * HID_, TPB), TPB, 0, stream>>>(b_ih, b_hh, bsum, h,
                                                           c);

  // ---- time loop (sequential LSTM scan)
  for (int t = 0; t < T_; ++t) {
    // xh[:, :300] = e_t ; xh[:, 2348:] = h
    dl_buildx_kernel<<<cdiv(B_ * (EMB_ + HID_), TPB), TPB, 0, stream>>>(
        emb_s, h, xh, t);

    // gate = sigmoid(h @ W_beta^T + b_beta); xh[:,300:2348] = gate * enc_mean
    {
      int waves = (B_ / 16) * (ENC_ / 32);  // MT=1 -> 256 waves
      dl_wmma_gemm_kernel<0, 1>
          <<<cdiv((long)waves * 32, TPB), TPB, 0, stream>>>(
              h, HID_, W_betaP, ENC_, HID_, b_beta, xh + EMB_, (size_t)XH_,
              enc_mean, dec_len, t);
    }

    // gates = [x|h] @ [W_ih|W_hh]^T + (b_ih + b_hh)
    {
      int waves = (B_ / 16) * (G4_ / 32);   // MT=1 -> 256 waves
      dl_wmma_gemm_kernel<1, 1>
          <<<cdiv((long)waves * 32, TPB), TPB, 0, stream>>>(
              xh, XH_, WcatP, G4_, XH_, bsum, gates, (size_t)G4_, nullptr,
              dec_len, t);
    }

    // LSTM cell + masked state update
    dl_cell_kernel<<<cdiv(B_ * HID_, TPB), TPB, 0, stream>>>(gates, dec_len, h,
                                                             c, h_new, t);

    // preds[:, t, :] = mask * (h_new @ W_fc^T + b_fc) -> d_out
    // MT=4: one wave covers all of M=64 -> W_fc read from L2 once per step.
    {
      int waves = VOCAB_ / 32;              // 625 waves
      dl_wmma_gemm_kernel<2, 4>
          <<<cdiv((long)waves * 32, TPB), TPB, 0, stream>>>(
              h_new, HID_, W_fcP, VOCAB_, HID_, b_fc,
              out_pred + (size_t)t * VOCAB_, (size_t)T_ * VOCAB_, nullptr,
              dec_len, t);
    }
  }
}